// CompactDocumentGNN_1047972020880
// MI455X (gfx1250) — compile-verified
//
#include <hip/hip_runtime.h>
#include <stdint.h>

typedef __attribute__((ext_vector_type(2))) float v2f;
typedef __attribute__((ext_vector_type(8))) float v8f;

#define N_NODES 150000
#define N_EDGES 2400000
#define F_IN    22
#define HID     32
#define N_CLS   6
#define BN_EPS  1e-5f
#define N_TILES (N_NODES / 16)   // 9375, exact

// ---------------------------------------------------------------------------
// 1) init: deg = 1.0 (self loop), stats[0..63] = 0
// ---------------------------------------------------------------------------
__global__ void init_kernel(float* __restrict__ deg, float* __restrict__ stats) {
    int i = blockIdx.x * blockDim.x + threadIdx.x;
    if (i < N_NODES) deg[i] = 1.0f;
    if (blockIdx.x == 0 && threadIdx.x < 64) stats[threadIdx.x] = 0.0f;
}

// ---------------------------------------------------------------------------
// 2) degree count over edge destinations (edge_index row 1)
// ---------------------------------------------------------------------------
__global__ void deg_count_kernel(const long long* __restrict__ edges,
                                 float* __restrict__ deg) {
    int e = blockIdx.x * blockDim.x + threadIdx.x;
    if (e < N_EDGES) {
        int dst = (int)edges[(size_t)N_EDGES + e];
        atomicAdd(&deg[dst], 1.0f);
    }
}

// ---------------------------------------------------------------------------
// 3) fused h2 = relu(x @ W_emb + b_emb) @ W_gcn   via V_WMMA_F32_16X16X4_F32
//    one wave per block, one 16-node tile per wave.
//    A fragment (16x4 f32): lanes 0-15 rows, VGPR pair holds K={0,1}|{2,3}.
//    C/D fragment (16x16 f32): VGPR r -> M = r + 8*(lane>=16), N = lane&15.
//    K-tail guards are value-masked (unconditional clamped-address loads +
//    v_cndmask) to avoid exec-predicated load blocks around the WMMAs.
// ---------------------------------------------------------------------------
__global__ __launch_bounds__(32) void embed_gemm_kernel(
    const float* __restrict__ x, const float* __restrict__ W_emb,
    const float* __restrict__ b_emb, const float* __restrict__ W_gcn,
    float* __restrict__ h2) {
    __shared__ float lds[16][33];          // stage-1 result, padded stride

    const int lane  = threadIdx.x & 31;
    const int tile  = blockIdx.x;          // 0 .. N_TILES-1, exact
    const int lrow  = lane & 15;
    const int col   = lane & 15;
    const int khalf = (lane >> 4) * 2;     // 0 for lanes 0-15, 2 for 16-31
    const int row   = tile * 16 + lrow;

    // ---- stage 1: E = relu(x @ W_emb + b_emb), K = 22 padded to 24 ----
    v8f c0 = {}; v8f c1 = {};
#pragma unroll
    for (int s = 0; s < 6; ++s) {
        int k0 = s * 4 + khalf;
        int k1 = k0 + 1;
        bool in0 = (k0 < F_IN);
        bool in1 = (k1 < F_IN);
        int kc0 = in0 ? k0 : 0;            // clamped (always-valid) indices
        int kc1 = in1 ? k1 : 0;
        // unconditional loads, value-masked afterwards
        float ax = x[(size_t)row * F_IN + kc0];
        float ay = x[(size_t)row * F_IN + kc1];
        float w00 = W_emb[(size_t)kc0 * HID + col];
        float w01 = W_emb[(size_t)kc1 * HID + col];
        float w10 = W_emb[(size_t)kc0 * HID + col + 16];
        float w11 = W_emb[(size_t)kc1 * HID + col + 16];
        v2f a, b0, b1;
        a.x  = in0 ? ax  : 0.0f;  a.y  = in1 ? ay  : 0.0f;
        b0.x = in0 ? w00 : 0.0f;  b0.y = in1 ? w01 : 0.0f;
        b1.x = in0 ? w10 : 0.0f;  b1.y = in1 ? w11 : 0.0f;
        c0 = __builtin_amdgcn_wmma_f32_16x16x4_f32(false, a, false, b0, (short)0, c0, false, false);
        c1 = __builtin_amdgcn_wmma_f32_16x16x4_f32(false, a, false, b1, (short)0, c1, false, false);
    }

    // bias + relu, stash row-major into LDS for the A-layout reload
    {
        float be0 = b_emb[col], be1 = b_emb[col + 16];
        int mbase = (lane >> 4) * 8;
#pragma unroll
        for (int r = 0; r < 8; ++r) {
            float e0 = c0[r] + be0; e0 = e0 > 0.0f ? e0 : 0.0f;
            float e1 = c1[r] + be1; e1 = e1 > 0.0f ? e1 : 0.0f;
            lds[mbase + r][col]      = e0;
            lds[mbase + r][col + 16] = e1;
        }
    }
    __syncthreads();   // fence the cross-lane LDS exchange (single wave)

    // ---- stage 2: H = E @ W_gcn, K = 32 (no tail, all loads unconditional) ----
    v8f d0 = {}; v8f d1 = {};
#pragma unroll
    for (int s = 0; s < 8; ++s) {
        int k0 = s * 4 + khalf;
        v2f a;
        a.x = lds[lrow][k0];
        a.y = lds[lrow][k0 + 1];
        v2f b0, b1;
        b0.x = W_gcn[(size_t)k0 * HID + col];
        b0.y = W_gcn[(size_t)(k0 + 1) * HID + col];
        b1.x = W_gcn[(size_t)k0 * HID + col + 16];
        b1.y = W_gcn[(size_t)(k0 + 1) * HID + col + 16];
        d0 = __builtin_amdgcn_wmma_f32_16x16x4_f32(false, a, false, b0, (short)0, d0, false, false);
        d1 = __builtin_amdgcn_wmma_f32_16x16x4_f32(false, a, false, b1, (short)0, d1, false, false);
    }
    {
        int mbase = (lane >> 4) * 8;
#pragma unroll
        for (int r = 0; r < 8; ++r) {
            int grow = tile * 16 + mbase + r;
            h2[(size_t)grow * HID + col]      = d0[r];
            h2[(size_t)grow * HID + col + 16] = d1[r];
        }
    }
}

// ---------------------------------------------------------------------------
// 4) agg[i,:] = h2[i,:] * (1/deg[i])  (self-loop: dis_i*dis_i) + b_gcn
//    8 threads per node, float4 each
// ---------------------------------------------------------------------------
__global__ void self_init_kernel(const float* __restrict__ h2,
                                 const float* __restrict__ deg,
                                 const float* __restrict__ b_gcn,
                                 float* __restrict__ agg) {
    int idx = blockIdx.x * blockDim.x + threadIdx.x;
    if (idx >= N_NODES * 8) return;
    int i = idx >> 3;
    int q = (idx & 7) * 4;
    float dinv = 1.0f / deg[i];
    float4 hv = *(const float4*)(h2 + (size_t)i * HID + q);
    float4 r;
    r.x = hv.x * dinv + b_gcn[q + 0];
    r.y = hv.y * dinv + b_gcn[q + 1];
    r.z = hv.z * dinv + b_gcn[q + 2];
    r.w = hv.w * dinv + b_gcn[q + 3];
    *(float4*)(agg + (size_t)i * HID + q) = r;
}

// ---------------------------------------------------------------------------
// 5) edge scatter: agg[dst] += h2[src] * rsqrt(deg[src]) * rsqrt(deg[dst])
//    8 threads per edge, float4 gather + 4 global float atomics
// ---------------------------------------------------------------------------
__global__ void scatter_kernel(const long long* __restrict__ edges,
                               const float* __restrict__ deg,
                               const float* __restrict__ h2,
                               float* __restrict__ agg) {
    long long idx = (long long)blockIdx.x * blockDim.x + threadIdx.x;
    if (idx >= (long long)N_EDGES * 8) return;
    int e = (int)(idx >> 3);
    int q = ((int)idx & 7) * 4;
    int src = (int)edges[e];
    int dst = (int)edges[(size_t)N_EDGES + e];
    float nrm = rsqrtf(deg[src]) * rsqrtf(deg[dst]);
    float4 hv = *(const float4*)(h2 + (size_t)src * HID + q);
    float* ap = agg + (size_t)dst * HID + q;
    atomicAdd(ap + 0, hv.x * nrm);
    atomicAdd(ap + 1, hv.y * nrm);
    atomicAdd(ap + 2, hv.z * nrm);
    atomicAdd(ap + 3, hv.w * nrm);
}

// ---------------------------------------------------------------------------
// 6) BN batch stats: stats[0..31] = sum per col, stats[32..63] = sumsq
// ---------------------------------------------------------------------------
__global__ __launch_bounds__(256) void bn_stats_kernel(
    const float* __restrict__ agg, float* __restrict__ stats) {
    __shared__ float ssum[256], ssq[256];
    int t = threadIdx.x;
    int col = t & 31;
    int rlane = t >> 5;  // 0..7
    float s = 0.0f, q = 0.0f;
    for (int i = blockIdx.x * 8 + rlane; i < N_NODES; i += gridDim.x * 8) {
        float v = agg[(size_t)i * HID + col];
        s += v; q += v * v;
    }
    ssum[t] = s; ssq[t] = q;
    __syncthreads();
    if (t < 32) {
#pragma unroll
        for (int j = 1; j < 8; ++j) { s += ssum[t + 32 * j]; q += ssq[t + 32 * j]; }
        atomicAdd(&stats[t], s);
        atomicAdd(&stats[32 + t], q);
    }
}

// ---------------------------------------------------------------------------
// 7) out = relu(BN(agg)) @ W_cls + b_cls   via V_WMMA_F32_16X16X4_F32
//    A fragments built on the fly with BN applied; N padded 6 -> 16.
//    W_cls/b_cls guards are value-masked (clamped index + select) so the
//    only exec-predicated path is the final store, after all WMMAs.
// ---------------------------------------------------------------------------
__global__ __launch_bounds__(32) void final_kernel(
    const float* __restrict__ agg, const float* __restrict__ stats,
    const float* __restrict__ gamma, const float* __restrict__ beta,
    const float* __restrict__ W_cls, const float* __restrict__ b_cls,
    float* __restrict__ out) {
    const int lane  = threadIdx.x & 31;
    const int tile  = blockIdx.x;
    const int lrow  = lane & 15;
    const int col   = lane & 15;
    const int khalf = (lane >> 4) * 2;
    const float invN = 1.0f / (float)N_NODES;
    const int grow = tile * 16 + lrow;
    const bool colv = (col < N_CLS);
    const int colc  = colv ? col : 0;      // clamped, always-valid column

    v8f d = {};
#pragma unroll
    for (int s = 0; s < 8; ++s) {
        int k0 = s * 4 + khalf;
        float m0 = stats[k0] * invN, m1 = stats[k0 + 1] * invN;
        float v0 = stats[32 + k0] * invN - m0 * m0;
        float v1 = stats[32 + k0 + 1] * invN - m1 * m1;
        float s0 = rsqrtf(v0 + BN_EPS) * gamma[k0];
        float s1 = rsqrtf(v1 + BN_EPS) * gamma[k0 + 1];
        float a0 = (agg[(size_t)grow * HID + k0]     - m0) * s0 + beta[k0];
        float a1 = (agg[(size_t)grow * HID + k0 + 1] - m1) * s1 + beta[k0 + 1];
        v2f a;
        a.x = a0 > 0.0f ? a0 : 0.0f;
        a.y = a1 > 0.0f ? a1 : 0.0f;
        // unconditional clamped loads, value-masked
        float w0 = W_cls[(size_t)k0 * N_CLS + colc];
        float w1 = W_cls[(size_t)(k0 + 1) * N_CLS + colc];
        v2f b;
        b.x = colv ? w0 : 0.0f;
        b.y = colv ? w1 : 0.0f;
        d = __builtin_amdgcn_wmma_f32_16x16x4_f32(false, a, false, b, (short)0, d, false, false);
    }
    float bc = b_cls[colc];
    if (colv) {
        int mbase = (lane >> 4) * 8;
#pragma unroll
        for (int r = 0; r < 8; ++r) {
            int orow = tile * 16 + mbase + r;
            out[(size_t)orow * N_CLS + col] = d[r] + bc;
        }
    }
}

// ---------------------------------------------------------------------------
// launch
// ---------------------------------------------------------------------------
extern "C" void kernel_launch(void* const* d_in, const int* in_sizes, int n_in,
                              void* d_out, int out_size, void* d_ws, size_t ws_size,
                              hipStream_t stream) {
    const float*     x     = (const float*)d_in[0];
    const long long* edges = (const long long*)d_in[1];  // int64 [2, E]
    const float*     W_emb = (const float*)d_in[2];
    const float*     b_emb = (const float*)d_in[3];
    const float*     W_gcn = (const float*)d_in[4];
    const float*     b_gcn = (const float*)d_in[5];
    const float*     gamma = (const float*)d_in[6];
    const float*     beta  = (const float*)d_in[7];
    const float*     W_cls = (const float*)d_in[8];
    const float*     b_cls = (const float*)d_in[9];
    float*           out   = (float*)d_out;

    // workspace layout (floats): deg[N] | stats[64] | h2[N*32] | agg[N*32]
    float* ws    = (float*)d_ws;
    float* deg   = ws;
    float* stats = ws + N_NODES;
    float* h2    = stats + 64;
    float* agg   = h2 + (size_t)N_NODES * HID;

    init_kernel<<<(N_NODES + 255) / 256, 256, 0, stream>>>(deg, stats);
    deg_count_kernel<<<(N_EDGES + 255) / 256, 256, 0, stream>>>(edges, deg);
    embed_gemm_kernel<<<N_TILES, 32, 0, stream>>>(x, W_emb, b_emb, W_gcn, h2);
    self_init_kernel<<<(N_NODES * 8 + 255) / 256, 256, 0, stream>>>(h2, deg, b_gcn, agg);
    scatter_kernel<<<(int)(((long long)N_EDGES * 8 + 255) / 256), 256, 0, stream>>>(edges, deg, h2, agg);
    bn_stats_kernel<<<256, 256, 0, stream>>>(agg, stats);
    final_kernel<<<N_TILES, 32, 0, stream>>>(agg, stats, gamma, beta, W_cls, b_cls, out);
}